// GraphSAGE_37056977830621
// MI455X (gfx1250) — compile-verified
//
#include <hip/hip_runtime.h>

// ---------------------------------------------------------------------------
// GraphSAGE on MI455X (gfx1250, wave32).
// GEMMs use V_WMMA_F32_16X16X4_F32 (full fp32 matrix pipe, exact precision).
// Scatter (mean-aggregation) uses L2 float atomics; all working data is
// L2-resident (51MB features, 26MB edges << 192MB L2).
// ---------------------------------------------------------------------------

typedef __attribute__((ext_vector_type(2))) float v2f;
typedef __attribute__((ext_vector_type(8))) float v8f;

#define N_NODES_C   100000
#define N_EDGES_C   1600000
#define D_C         128
#define N_CLASSES_C 64

// ---------------------------------------------------------------------------
// Utility: zero a float buffer (d_ws is poisoned with 0xAA by the harness).
__global__ __launch_bounds__(256) void zero_kernel(float* __restrict__ p, int n) {
    int i = blockIdx.x * 256 + threadIdx.x;
    if (i < n) p[i] = 0.0f;
}

// ---------------------------------------------------------------------------
// Degree accumulation: one thread per edge.
__global__ __launch_bounds__(256) void degree_kernel(const long long* __restrict__ dst,
                                                     float* __restrict__ deg, int n_edges) {
    int e = blockIdx.x * 256 + threadIdx.x;
    if (e < n_edges) atomicAdd(&deg[(int)dst[e]], 1.0f);
}

__global__ __launch_bounds__(256) void invdeg_kernel(float* __restrict__ deg, int n) {
    int i = blockIdx.x * 256 + threadIdx.x;
    if (i < n) {
        float d = deg[i];
        deg[i] = 1.0f / (d < 1.0f ? 1.0f : d);
    }
}

// ---------------------------------------------------------------------------
// Neighbor scatter-sum: one wave (32 lanes) per edge, 4 channels per lane.
// agg[dst] += h[src].  204.8M f32 atomics/layer -> L2 atomic units.
__global__ __launch_bounds__(256) void scatter_kernel(const float* __restrict__ h,
                                                      const long long* __restrict__ src,
                                                      const long long* __restrict__ dst,
                                                      float* __restrict__ agg, int n_edges) {
    long long gid = (long long)blockIdx.x * 256 + threadIdx.x;
    int e    = (int)(gid >> 5);   // uniform within a wave
    int lane = (int)(gid & 31);
    if (e >= n_edges) return;
    int s = (int)src[e];
    int d = (int)dst[e];
    const float4 v = *(const float4*)(h + (size_t)s * D_C + lane * 4);
    float* o = agg + (size_t)d * D_C + lane * 4;
    atomicAdd(o + 0, v.x);
    atomicAdd(o + 1, v.y);
    atomicAdd(o + 2, v.z);
    atomicAdd(o + 3, v.w);
}

// ---------------------------------------------------------------------------
// SAGE layer GEMM:
//   hout = relu( (agg * inv_deg) @ Wl^T + bl + h @ Wr^T )
// Block = 256 threads = 8 waves. Block handles 16 nodes; wave w handles the
// 16-column tile [w*16, w*16+16). K = 128 -> 32 k-steps of the fp32 WMMA,
// two independent accumulator chains (Wl path / Wr path) for ILP.
//
// fp32 WMMA fragment addressing (ISA 7.12.2):
//   A: lane l -> row m=l%16, float2 at K = kb + 2*(l/16)
//   B[k][j] = W[j][k]: lane l -> col j, float2 at K = kb + 2*(l/16)  (row j of W)
//   C/D: acc[r] -> out[row rbase + r + 8*(l/16)][col tile*16 + l%16]
__global__ __launch_bounds__(256) void sage_gemm_kernel(const float* __restrict__ h,
                                                        const float* __restrict__ agg,
                                                        const float* __restrict__ inv_deg,
                                                        const float* __restrict__ Wl,
                                                        const float* __restrict__ bl,
                                                        const float* __restrict__ Wr,
                                                        float* __restrict__ hout) {
    const int wave = threadIdx.x >> 5;   // 0..7 -> N tile
    const int lane = threadIdx.x & 31;
    const int m    = lane & 15;
    const int half = lane >> 4;
    const int rbase = blockIdx.x * 16;   // node tile base (N_NODES % 16 == 0)
    const int row   = rbase + m;         // node row this lane loads A for
    const int col   = wave * 16 + m;     // output column this lane loads B for

    const float id = inv_deg[row];
    const float* __restrict__ arow  = agg + (size_t)row * D_C;
    const float* __restrict__ hrow  = h   + (size_t)row * D_C;
    const float* __restrict__ wlrow = Wl  + (size_t)col * D_C;
    const float* __restrict__ wrrow = Wr  + (size_t)col * D_C;

    const float bj = bl[col];
    v8f acc1, acc2;
#pragma unroll
    for (int r = 0; r < 8; ++r) { acc1[r] = bj; acc2[r] = 0.0f; }

#pragma unroll 8
    for (int kb = 0; kb < D_C; kb += 4) {
        const int ko = kb + 2 * half;
        v2f a1 = *(const v2f*)(arow + ko);
        a1 *= id;                                   // fused mean scaling
        v2f b1 = *(const v2f*)(wlrow + ko);
        acc1 = __builtin_amdgcn_wmma_f32_16x16x4_f32(false, a1, false, b1,
                                                     (short)0, acc1, false, false);
        v2f a2 = *(const v2f*)(hrow + ko);
        v2f b2 = *(const v2f*)(wrrow + ko);
        acc2 = __builtin_amdgcn_wmma_f32_16x16x4_f32(false, a2, false, b2,
                                                     (short)0, acc2, false, false);
    }

#pragma unroll
    for (int r = 0; r < 8; ++r) {
        float v = acc1[r] + acc2[r];
        v = v < 0.0f ? 0.0f : v;                    // ReLU
        hout[(size_t)(rbase + r + 8 * half) * D_C + col] = v;
    }
}

// ---------------------------------------------------------------------------
// Output head: out = h @ Wout^T + bout.  64 classes -> 4 waves per block.
__global__ __launch_bounds__(128) void out_gemm_kernel(const float* __restrict__ h,
                                                       const float* __restrict__ Wout,
                                                       const float* __restrict__ bout,
                                                       float* __restrict__ out) {
    const int wave = threadIdx.x >> 5;   // 0..3 -> class tile
    const int lane = threadIdx.x & 31;
    const int m    = lane & 15;
    const int half = lane >> 4;
    const int rbase = blockIdx.x * 16;
    const int row   = rbase + m;
    const int col   = wave * 16 + m;

    const float* __restrict__ hrow = h    + (size_t)row * D_C;
    const float* __restrict__ wrow = Wout + (size_t)col * D_C;

    const float bj = bout[col];
    v8f acc;
#pragma unroll
    for (int r = 0; r < 8; ++r) acc[r] = bj;

#pragma unroll 8
    for (int kb = 0; kb < D_C; kb += 4) {
        const int ko = kb + 2 * half;
        v2f a = *(const v2f*)(hrow + ko);
        v2f b = *(const v2f*)(wrow + ko);
        acc = __builtin_amdgcn_wmma_f32_16x16x4_f32(false, a, false, b,
                                                    (short)0, acc, false, false);
    }

#pragma unroll
    for (int r = 0; r < 8; ++r)
        out[(size_t)(rbase + r + 8 * half) * N_CLASSES_C + col] = acc[r];
}

// ---------------------------------------------------------------------------
extern "C" void kernel_launch(void* const* d_in, const int* in_sizes, int n_in,
                              void* d_out, int out_size, void* d_ws, size_t ws_size,
                              hipStream_t stream) {
    (void)in_sizes; (void)n_in; (void)out_size; (void)ws_size;

    const float*     x    = (const float*)d_in[0];                 // [N,128]
    const long long* ei   = (const long long*)d_in[1];             // [2,E] int64
    const float*     Wl   = (const float*)d_in[2];                 // [3,128,128]
    const float*     bl   = (const float*)d_in[3];                 // [3,128]
    const float*     Wr   = (const float*)d_in[4];                 // [3,128,128]
    const float*     Wout = (const float*)d_in[5];                 // [64,128]
    const float*     bout = (const float*)d_in[6];                 // [64]
    float*           out  = (float*)d_out;                        // [N,64]

    const long long* src = ei;
    const long long* dst = ei + N_EDGES_C;

    // Workspace carve-up (all fp32):
    float* inv_deg = (float*)d_ws;                                 // N
    float* agg     = inv_deg + N_NODES_C;                          // N*128
    float* hA      = agg + (size_t)N_NODES_C * D_C;                // N*128
    float* hB      = hA  + (size_t)N_NODES_C * D_C;                // N*128

    const int feat_elems = N_NODES_C * D_C;                        // 12.8M

    // Degrees -> inverse degrees (deg accumulated in inv_deg buffer).
    zero_kernel<<<(N_NODES_C + 255) / 256, 256, 0, stream>>>(inv_deg, N_NODES_C);
    degree_kernel<<<(N_EDGES_C + 255) / 256, 256, 0, stream>>>(dst, inv_deg, N_EDGES_C);
    invdeg_kernel<<<(N_NODES_C + 255) / 256, 256, 0, stream>>>(inv_deg, N_NODES_C);

    const int gemm_blocks    = N_NODES_C / 16;                           // 6250
    const int scatter_blocks = (int)(((long long)N_EDGES_C * 32 + 255) / 256);

    const float* hin  = x;
    float*       hcur = hA;
    float*       hnxt = hB;

    for (int layer = 0; layer < 3; ++layer) {
        zero_kernel<<<(feat_elems + 255) / 256, 256, 0, stream>>>(agg, feat_elems);
        scatter_kernel<<<scatter_blocks, 256, 0, stream>>>(hin, src, dst, agg, N_EDGES_C);
        sage_gemm_kernel<<<gemm_blocks, 256, 0, stream>>>(
            hin, agg, inv_deg,
            Wl + (size_t)layer * D_C * D_C, bl + (size_t)layer * D_C,
            Wr + (size_t)layer * D_C * D_C, hcur);
        hin = hcur;
        float* t = hcur; hcur = hnxt; hnxt = t;
    }

    out_gemm_kernel<<<gemm_blocks, 128, 0, stream>>>(hin, Wout, bout, out);
}